// DWT_Features_67035849556150
// MI455X (gfx1250) — compile-verified
//
#include <hip/hip_runtime.h>
#include <hip/hip_bf16.h>
#include <cstddef>

typedef __attribute__((ext_vector_type(2))) float v2f;
typedef __attribute__((ext_vector_type(8))) float v8f;

// db4 analysis filters (pywt dec_lo / dec_hi)
__device__ __constant__ float g_dec_lo[8] = {
    -0.010597401784997278f,  0.032883011666982945f,  0.030841381835986965f,
    -0.18703481171888114f,  -0.02798376941698385f,   0.6308807679295904f,
     0.7148465705525415f,    0.23037781330885523f};
__device__ __constant__ float g_dec_hi[8] = {
    -0.23037781330885523f,   0.7148465705525415f,   -0.6308807679295904f,
    -0.02798376941698385f,   0.18703481171888114f,   0.030841381835986965f,
    -0.032883011666982945f, -0.010597401784997278f};

// ---------------------------------------------------------------------------
// Kernel 1: build the 84x64 linear DWT matrix M (M[d*64+t]) by running the
// 3-level db4 analysis bank (reflect padding, stride 2, reversed filters,
// i.e. out[o] = sum_j xp[2o+j]*DEC[7-j]) on each unit basis vector e_t.
// ---------------------------------------------------------------------------
__global__ void build_dwt_matrix(float* __restrict__ Mout) {
    const int t = threadIdx.x;
    if (t >= 64) return;

    float cur[64];
    for (int i = 0; i < 64; ++i) cur[i] = (i == t) ? 1.0f : 0.0f;

    const int outs[3]   = {35, 21, 14};   // level output lengths (N=64->35->21->14)
    const int hibase[3] = {14, 49, 70};   // hi-coeff destinations in the 84-vector
    int N = 64;
    for (int lev = 0; lev < 3; ++lev) {
        float lo[35];
        const int On = outs[lev];
        for (int o = 0; o < On; ++o) {
            float sl = 0.0f, sh = 0.0f;
            for (int j = 0; j < 8; ++j) {
                int idx = 2 * o + j - 6;              // left pad is always 6
                if (idx < 0)   idx = -idx;            // reflect (no edge repeat)
                if (idx >= N)  idx = 2 * (N - 1) - idx;
                const float v = cur[idx];
                sl += v * g_dec_lo[7 - j];
                sh += v * g_dec_hi[7 - j];
            }
            lo[o] = sl;
            Mout[(hibase[lev] + o) * 64 + t] = sh;    // hi coeffs
        }
        for (int o = 0; o < On; ++o) cur[o] = lo[o];
        N = On;
    }
    for (int o = 0; o < 14; ++o) Mout[o * 64 + t] = cur[o];  // final lo coeffs
}

// ---------------------------------------------------------------------------
// Kernel 2: fold M into conv_w:
//   W_eff[k][t*64+hw] = sum_{d<84} M[d*64+t] * conv_w[k*5376 + d*64 + hw]
// 16*4096 = 65536 outputs, 84 MACs each (trivial).
// ---------------------------------------------------------------------------
__global__ __launch_bounds__(256)
void build_weff(const float* __restrict__ conv_w, const float* __restrict__ M,
                float* __restrict__ Weff) {
    const int idx = blockIdx.x * 256 + threadIdx.x;   // 0..65535
    const int k   = idx >> 12;
    const int rem = idx & 4095;
    const int t   = rem >> 6;
    const int hw  = rem & 63;

    const float* wk = conv_w + (size_t)k * 5376 + hw;
    const float* Mc = M + t;
    float s = 0.0f;
#pragma unroll 4
    for (int d = 0; d < 84; ++d) s += Mc[d * 64] * wk[d * 64];
    Weff[idx] = s;
}

// ---------------------------------------------------------------------------
// Kernel 3: the GEMM. Each wave32 computes one 16x16 tile of
//   feat[16384,16] = x[16384,4096] * W_eff^T[4096,16]  (+bias, LeakyReLU)
// using V_WMMA_F32_16X16X4_F32, marching K in steps of 4.
//
// A fragment (16x4, MxK): lanes 0-15 row M=lane, VGPR0=K0 VGPR1=K1;
//                         lanes 16-31 row M=lane-16, VGPR0=K2 VGPR1=K3.
//   -> one global_load_b64 per lane at x[(row0+(l&15))*4096 + s + 2*(l>>4)]
// B fragment (4x16, KxN): lanes = N (channel), same K split per lane half.
//   -> one global_load_b64 per lane at W[(l&15)*4096 + s + 2*(l>>4)]
// Two accumulator chains cover the WMMA->WMMA RAW hazard latency.
// ---------------------------------------------------------------------------
__global__ __launch_bounds__(256)
void dwt_gemm_wmma(const float* __restrict__ x, const float* __restrict__ Weff,
                   const float* __restrict__ bias, float* __restrict__ out) {
    const int lane = threadIdx.x & 31;
    const int wave = threadIdx.x >> 5;
    const int tile = blockIdx.x * 8 + wave;    // 0..1023
    const int row0 = tile * 16;
    const int nsel = lane & 15;                 // M row (A) / channel N (B)
    const int koff = (lane >> 4) * 2;           // 0 or 2 within each K=4 chunk

    const float* aptr = x    + (size_t)(row0 + nsel) * 4096 + koff;
    const float* bptr = Weff + (size_t)nsel * 4096 + koff;

    v8f acc0 = {};
    v8f acc1 = {};
#pragma unroll 4
    for (int c = 0; c < 1024; c += 2) {
        const v2f a0 = *(const v2f*)(aptr + 4 * c);
        const v2f b0 = *(const v2f*)(bptr + 4 * c);
        const v2f a1 = *(const v2f*)(aptr + 4 * c + 4);
        const v2f b1 = *(const v2f*)(bptr + 4 * c + 4);
        acc0 = __builtin_amdgcn_wmma_f32_16x16x4_f32(
            false, a0, false, b0, (short)0, acc0, false, false);
        acc1 = __builtin_amdgcn_wmma_f32_16x16x4_f32(
            false, a1, false, b1, (short)0, acc1, false, false);
    }
    const v8f acc = acc0 + acc1;

    // D layout: VGPR r -> (M=r, N=lane) for lanes 0-15, (M=r+8, N=lane-16) hi.
    const float kb   = bias[nsel];
    const int   mrow = row0 + (lane >> 4) * 8;
#pragma unroll
    for (int r = 0; r < 8; ++r) {
        float f = acc[r] + kb;
        f = (f >= 0.0f) ? f : 1e-3f * f;        // LeakyReLU(0.001)
        out[(size_t)(mrow + r) * 16 + nsel] = f;
    }
}

extern "C" void kernel_launch(void* const* d_in, const int* in_sizes, int n_in,
                              void* d_out, int out_size, void* d_ws, size_t ws_size,
                              hipStream_t stream) {
    const float* x      = (const float*)d_in[0];   // [16384,1,64,8,8]
    const float* conv_w = (const float*)d_in[1];   // [16,1,84,8,8]
    const float* conv_b = (const float*)d_in[2];   // [16]
    float*       out    = (float*)d_out;           // [16384,16]

    float* Weff = (float*)d_ws;                    // 16*4096 floats = 256 KB
    float* Mbuf = Weff + 16 * 4096;                // 84*64 floats   = 21 KB

    build_dwt_matrix<<<1, 64, 0, stream>>>(Mbuf);
    build_weff<<<256, 256, 0, stream>>>(conv_w, Mbuf, Weff);
    dwt_gemm_wmma<<<128, 256, 0, stream>>>(x, Weff, conv_b, out);
}